// GCN_51634096833134
// MI455X (gfx1250) — compile-verified
//
#include <hip/hip_runtime.h>
#include <math.h>

// ---------------------------------------------------------------------------
// GCN layer for MI455X (gfx1250, wave32):
//   h   = x @ W.T + b                      (fp32 WMMA 16x16x4)
//   agg = segment_sum(h[src] * norm, dst)  (L2-resident gather + HW fp32 atomics)
//   out = gelu(agg @ W.T + b)              (fp32 WMMA + exact-erf GELU)
// ---------------------------------------------------------------------------

typedef __attribute__((ext_vector_type(2))) float v2f;
typedef __attribute__((ext_vector_type(8))) float v8f;

#define HIDDEN 128

// ---------------- zero workspace (agg + integer degree) --------------------
__global__ __launch_bounds__(256) void zero_ws_kernel(float* __restrict__ agg,
                                                      int* __restrict__ degi,
                                                      long total, int n_nodes) {
  long i = (long)blockIdx.x * blockDim.x + threadIdx.x;
  if (i < total) agg[i] = 0.0f;
  if (i < n_nodes) degi[i] = 0;
}

// ---------------- in-degree via integer atomics ----------------------------
__global__ __launch_bounds__(256) void degree_kernel(const int* __restrict__ dst,
                                                     int* __restrict__ degi, int E) {
  int e = blockIdx.x * blockDim.x + threadIdx.x;
  if (e < E) atomicAdd(&degi[dst[e]], 1);
}

// ---------------- deg^-0.5 -------------------------------------------------
__global__ __launch_bounds__(256) void dinv_kernel(const int* __restrict__ degi,
                                                   float* __restrict__ dinv, int n) {
  int i = blockIdx.x * blockDim.x + threadIdx.x;
  if (i < n) dinv[i] = rsqrtf((float)degi[i]);  // in-degree >= 1 guaranteed
}

// ---------------- fp32 WMMA GEMM:  Y = X @ W.T + b  (optional exact GELU) --
// grid.x = nrows/16 (nrows is a multiple of 16: 50000 = 3125*16)
// block  = 256 threads = 8 waves; wave w owns output columns [16w, 16w+16)
__global__ __launch_bounds__(256) void gemm_wmma_f32_kernel(
    const float* __restrict__ X, const float* __restrict__ W,
    const float* __restrict__ bias, float* __restrict__ Y, int apply_gelu) {
  const int wave = threadIdx.x >> 5;   // 0..7  -> 16-col tile
  const int lane = threadIdx.x & 31;
  const int lm   = lane & 15;
  const int half = lane >> 4;          // 0 or 1
  const int row0 = blockIdx.x * 16;
  const int col  = wave * 16 + lm;     // this lane's output column (B/C N index)

  // ISA §7.12.2 (32-bit 16x4 A): lane lm holds row M=lm; VGPR0/1 = K = 2*half, 2*half+1
  const float* arow = X + (long)(row0 + lm) * HIDDEN + 2 * half;
  // B = W^T (KxN): B[k][col] = W[col][k] -> contiguous float2 along W's row
  const float* brow = W + (long)col * HIDDEN + 2 * half;

  v8f c = {};
#pragma unroll
  for (int k0 = 0; k0 < HIDDEN; k0 += 4) {
    v2f a = *(const v2f*)(arow + k0);
    v2f bf = *(const v2f*)(brow + k0);
    // D = A(16x4) * B(4x16) + C ;  emits v_wmma_f32_16x16x4_f32
    c = __builtin_amdgcn_wmma_f32_16x16x4_f32(
        /*neg_a=*/false, a, /*neg_b=*/false, bf,
        /*c_mod=*/(short)0, c, /*reuse_a=*/false, /*reuse_b=*/false);
  }

  const float bc = bias[col];
#pragma unroll
  for (int v = 0; v < 8; ++v) {
    // C/D layout: VGPR v -> tile row (v + 8*half); lane lm -> tile col
    int r = row0 + v + 8 * half;
    float val = c[v] + bc;
    if (apply_gelu) {
      // exact (erf-based) GELU, matching torch default / jax approximate=False
      val = 0.5f * val * (1.0f + erff(val * 0.70710678118654752440f));
    }
    Y[(long)r * HIDDEN + col] = val;
  }
}

// ---------------- SpMM: agg[dst] += h[src] * norm --------------------------
// One wave per edge; each lane moves a float4 chunk (32 lanes * 4 = 128).
// h and agg both fit in the 192 MB L2 -> gather + atomics stay on-chip.
__global__ __launch_bounds__(256) void spmm_scatter_kernel(
    const float* __restrict__ h, const int* __restrict__ src,
    const int* __restrict__ dst, const float* __restrict__ dinv,
    float* __restrict__ agg, int E) {
  int e = blockIdx.x * 8 + (threadIdx.x >> 5);  // 8 edges per 256-thread block
  if (e >= E) return;
  const int lane = threadIdx.x & 31;
  const int s = src[e];
  const int d = dst[e];
  const float norm = dinv[s] * dinv[d];

  const float4 v = ((const float4*)(h + (long)s * HIDDEN))[lane];
  float* arow = agg + (long)d * HIDDEN + lane * 4;
  // unsafeAtomicAdd -> hardware global_atomic_add_f32 (no CAS loop)
  unsafeAtomicAdd(arow + 0, v.x * norm);
  unsafeAtomicAdd(arow + 1, v.y * norm);
  unsafeAtomicAdd(arow + 2, v.z * norm);
  unsafeAtomicAdd(arow + 3, v.w * norm);
}

// ---------------------------------------------------------------------------
extern "C" void kernel_launch(void* const* d_in, const int* in_sizes, int n_in,
                              void* d_out, int out_size, void* d_ws, size_t ws_size,
                              hipStream_t stream) {
  const float* x    = (const float*)d_in[0];   // [N, 128]
  const float* W    = (const float*)d_in[1];   // [128, 128]
  const float* bias = (const float*)d_in[2];   // [128]
  const int*   ei   = (const int*)d_in[3];     // [2, E] row0=src, row1=dst

  const int N = in_sizes[0] / HIDDEN;          // 50000 (multiple of 16)
  const int E = in_sizes[3] / 2;               // 800000
  const int* src = ei;
  const int* dst = ei + E;

  // workspace layout: h[N*128] | agg[N*128] | degi[N] (int) | dinv[N]
  float* h    = (float*)d_ws;
  float* agg  = h + (long)N * HIDDEN;
  int*   degi = (int*)(agg + (long)N * HIDDEN);
  float* dinv = (float*)(degi + N);

  const long totz = (long)N * HIDDEN;
  zero_ws_kernel<<<(unsigned)((totz + 255) / 256), 256, 0, stream>>>(agg, degi, totz, N);
  degree_kernel<<<(E + 255) / 256, 256, 0, stream>>>(dst, degi, E);
  dinv_kernel<<<(N + 255) / 256, 256, 0, stream>>>(degi, dinv, N);

  // GEMM1: h = x @ W.T + b
  gemm_wmma_f32_kernel<<<N / 16, 256, 0, stream>>>(x, W, bias, h, /*gelu=*/0);

  // SpMM scatter-add
  spmm_scatter_kernel<<<(E + 7) / 8, 256, 0, stream>>>(h, src, dst, dinv, agg, E);

  // GEMM2 + GELU: out = gelu(agg @ W.T + b)
  gemm_wmma_f32_kernel<<<N / 16, 256, 0, stream>>>(agg, W, bias, (float*)d_out,
                                                   /*gelu=*/1);
}